// GatingNeuralAdaptiveBias_82291573391764
// MI455X (gfx1250) — compile-verified
//
#include <hip/hip_runtime.h>
#include <math.h>

typedef float v2f __attribute__((ext_vector_type(2)));
typedef float v8f __attribute__((ext_vector_type(8)));

#define WAVES 2
#define THREADS (WAVES * 32)
#define NPIX (8 * 256 * 256)

// padded row strides (dwords) -> conflict-free banks, 8B-aligned (even)
#define XS 34   // features: ch0[0..15] | ch1[16..31] | pad
#define HS 66   // hidden:   [0..63] | pad
#define ES 130  // encoded:  e0[0..63] | e1[64..127] | pad
#define W1S 18  // W1^T row stride

__device__ __forceinline__ float fast_rcp(float x) { return __builtin_amdgcn_rcpf(x); }
__device__ __forceinline__ float fast_rsq(float x) { return __builtin_amdgcn_rsqf(x); }

__device__ __forceinline__ float silu_f(float x) {
    // x * sigmoid(x) via v_exp_f32 + v_rcp_f32 (no IEEE divide expansion)
    return x * fast_rcp(1.0f + __expf(-x));
}

__device__ __forceinline__ v8f wmma4(v2f a, v2f b, v8f c) {
    // D = A(16x4,f32) * B(4x16,f32) + C(16x16,f32), full fp32 matrix core
    return __builtin_amdgcn_wmma_f32_16x16x4_f32(
        false, a, false, b, (short)0, c, false, false);
}

__global__ __launch_bounds__(THREADS)
void GatingNeuralAdaptiveBias_kernel(
    const float* __restrict__ coords,      // (8,256,2)
    const float* __restrict__ cost_mat,    // (8,256,256)
    const float* __restrict__ log_scale,   // (2,)
    const float* __restrict__ W1,          // (9,64)
    const float* __restrict__ b1,          // (64,)
    const float* __restrict__ W2,          // (64,64)
    const float* __restrict__ b2,          // (64,)
    const float* __restrict__ film_gamma,  // (2,64)
    const float* __restrict__ film_beta,   // (2,64)
    const float* __restrict__ Wg1,         // (128,64)
    const float* __restrict__ bg1,         // (64,)
    const float* __restrict__ Wg2,         // (64,2)
    const float* __restrict__ bg2,         // (2,)
    const float* __restrict__ gate_temp,   // (1,)
    const float* __restrict__ ln_gamma,    // (64,)
    const float* __restrict__ ln_beta,     // (64,)
    const float* __restrict__ Wo,          // (64,1)
    const float* __restrict__ bo,          // (1,)
    float* __restrict__ out)               // (8,256,256)
{
    __shared__ float sW1t[64 * W1S];           // 4.5 KB zero-padded W1^T [n][k]
    __shared__ float sX[WAVES][32][XS];        // 8.5 KB
    __shared__ float sH[WAVES][32][HS];        // 16.5 KB
    __shared__ float sE[WAVES][32][ES];        // 32.5 KB

    const int tid = threadIdx.x;

    // ---- cooperative zero-padded W1^T staging ----
    for (int idx = tid; idx < 64 * 16; idx += THREADS) {
        const int n = idx >> 4;
        const int k = idx & 15;
        sW1t[n * W1S + k] = (k < 9) ? W1[k * 64 + n] : 0.0f;
    }
    __syncthreads();

    const int w    = tid >> 5;
    const int lane = tid & 31;
    const int rl   = lane & 15;   // N%16 for B/C frags, row within M-tile
    const int half = lane >> 4;   // K-pair / M-block selector

    float (*X)[XS] = sX[w];
    float (*H)[HS] = sH[w];
    float (*E)[ES] = sE[w];

    // ---- per-pixel features: one lane <-> one pixel ----
    const int p0   = blockIdx.x * (WAVES * 32) + w * 32;
    const int p    = p0 + lane;
    const int bidx = p >> 16;
    const int i    = (p >> 8) & 255;
    const int j    = p & 255;

    const float c   = cost_mat[p];
    const float dix = coords[(bidx * 256 + i) * 2 + 0] - coords[(bidx * 256 + j) * 2 + 0];
    const float diy = coords[(bidx * 256 + i) * 2 + 1] - coords[(bidx * 256 + j) * 2 + 1];
    const float ang = atan2f(diy, dix);
    const float s0  = __expf(log_scale[0]);
    const float s1  = __expf(log_scale[1]);

    float f0[16], f1[16];
#pragma unroll
    for (int k = 0; k < 16; k++) { f0[k] = 0.0f; f1[k] = 0.0f; }
    f0[0] = c * s0;
    f1[0] = ang * s1;
#pragma unroll
    for (int fq = 0; fq < 4; fq++) {
        const float fr = (float)(1 << fq);   // 1,2,4,8
        float ss, cc;
        __sincosf(c * fr, &ss, &cc);
        f0[1 + 2 * fq] = ss * s0;
        f0[2 + 2 * fq] = cc * s0;
        __sincosf(ang * fr, &ss, &cc);
        f1[1 + 2 * fq] = ss * s1;
        f1[2 + 2 * fq] = cc * s1;
    }
#pragma unroll
    for (int k = 0; k < 16; k++) {
        X[lane][k]      = f0[k];
        X[lane][16 + k] = f1[k];
    }
    __asm volatile("" ::: "memory");   // same-wave DS ops are HW in-order

    // ---- two-channel encoder ----
#pragma unroll
    for (int ch = 0; ch < 2; ch++) {
        // layer1: two 16-row tiles share every B fragment
        v8f acc[2][4];
#pragma unroll
        for (int nt = 0; nt < 4; nt++) {
            const float bv = b1[rl + nt * 16];
            v8f t = {bv, bv, bv, bv, bv, bv, bv, bv};
            acc[0][nt] = t; acc[1][nt] = t;
        }
#pragma unroll
        for (int k4 = 0; k4 < 16; k4 += 4) {
            const int kk = k4 + 2 * half;
            v2f bf[4];
#pragma unroll
            for (int nt = 0; nt < 4; nt++)
                bf[nt] = *(const v2f*)&sW1t[(rl + nt * 16) * W1S + kk];
#pragma unroll
            for (int mt = 0; mt < 2; mt++) {
                v2f a = *(const v2f*)&X[mt * 16 + rl][ch * 16 + kk];
#pragma unroll
                for (int nt = 0; nt < 4; nt++)
                    acc[mt][nt] = wmma4(a, bf[nt], acc[mt][nt]);
            }
        }
        // silu -> H
#pragma unroll
        for (int mt = 0; mt < 2; mt++)
#pragma unroll
            for (int nt = 0; nt < 4; nt++) {
                const int n = rl + nt * 16;
#pragma unroll
                for (int e = 0; e < 8; e++)
                    H[mt * 16 + e + 8 * half][n] = silu_f(acc[mt][nt][e]);
            }
        __asm volatile("" ::: "memory");

        // layer2: [32x64] @ [64x64]
        v8f acc2[2][4];
#pragma unroll
        for (int nt = 0; nt < 4; nt++) {
            const float bv = b2[rl + nt * 16];
            v8f t = {bv, bv, bv, bv, bv, bv, bv, bv};
            acc2[0][nt] = t; acc2[1][nt] = t;
        }
#pragma unroll 4
        for (int k4 = 0; k4 < 64; k4 += 4) {
            const int kk = k4 + 2 * half;
            v2f bf[4];
#pragma unroll
            for (int nt = 0; nt < 4; nt++) {
                const int n = rl + nt * 16;
                bf[nt].x = W2[kk * 64 + n];
                bf[nt].y = W2[(kk + 1) * 64 + n];
            }
#pragma unroll
            for (int mt = 0; mt < 2; mt++) {
                v2f a = *(const v2f*)&H[mt * 16 + rl][kk];
#pragma unroll
                for (int nt = 0; nt < 4; nt++)
                    acc2[mt][nt] = wmma4(a, bf[nt], acc2[mt][nt]);
            }
        }
        // FiLM -> E[:, ch*64 + n]
#pragma unroll
        for (int mt = 0; mt < 2; mt++)
#pragma unroll
            for (int nt = 0; nt < 4; nt++) {
                const int n = rl + nt * 16;
                const float g  = film_gamma[ch * 64 + n];
                const float bt = film_beta[ch * 64 + n];
#pragma unroll
                for (int e = 0; e < 8; e++)
                    E[mt * 16 + e + 8 * half][ch * 64 + n] = acc2[mt][nt][e] * g + bt;
            }
        __asm volatile("" ::: "memory");
    }

    // ---- gate: [32x128] @ [128x64] ----
    v8f acc3[2][4];
#pragma unroll
    for (int nt = 0; nt < 4; nt++) {
        const float bv = bg1[rl + nt * 16];
        v8f t = {bv, bv, bv, bv, bv, bv, bv, bv};
        acc3[0][nt] = t; acc3[1][nt] = t;
    }
#pragma unroll 4
    for (int k4 = 0; k4 < 128; k4 += 4) {
        const int kk = k4 + 2 * half;
        v2f bf[4];
#pragma unroll
        for (int nt = 0; nt < 4; nt++) {
            const int n = rl + nt * 16;
            bf[nt].x = Wg1[kk * 64 + n];
            bf[nt].y = Wg1[(kk + 1) * 64 + n];
        }
#pragma unroll
        for (int mt = 0; mt < 2; mt++) {
            v2f a = *(const v2f*)&E[mt * 16 + rl][kk];
#pragma unroll
            for (int nt = 0; nt < 4; nt++)
                acc3[mt][nt] = wmma4(a, bf[nt], acc3[mt][nt]);
        }
    }
    // silu -> gate hidden (reuse H)
#pragma unroll
    for (int mt = 0; mt < 2; mt++)
#pragma unroll
        for (int nt = 0; nt < 4; nt++) {
            const int n = rl + nt * 16;
#pragma unroll
            for (int e = 0; e < 8; e++)
                H[mt * 16 + e + 8 * half][n] = silu_f(acc3[mt][nt][e]);
        }
    __asm volatile("" ::: "memory");

    // ---- tails: each lane owns pixel `p` (row `lane`), zero divergence ----
    float pl0 = 0.0f, pl1 = 0.0f;
#pragma unroll
    for (int f = 0; f < 64; f++) {
        const float hv = H[lane][f];
        pl0 += hv * Wg2[f * 2 + 0];
        pl1 += hv * Wg2[f * 2 + 1];
    }
    const float l0 = pl0 + bg2[0];
    const float l1 = pl1 + bg2[1];
    const float invT = __expf(-gate_temp[0]);
    const float z0 = l0 * invT, z1 = l1 * invT;
    const float zm = fmaxf(z0, z1);
    const float ew0 = __expf(z0 - zm), ew1 = __expf(z1 - zm);
    const float w0 = ew0 * fast_rcp(ew0 + ew1);
    const float w1 = 1.0f - w0;

    float s = 0.0f, sq = 0.0f;
#pragma unroll
    for (int f = 0; f < 64; f++) {
        const float v = w0 * E[lane][f] + w1 * E[lane][64 + f];
        s  += v;
        sq += v * v;
    }
    const float mu   = s * (1.0f / 64.0f);
    const float var  = sq * (1.0f / 64.0f) - mu * mu;
    const float rstd = fast_rsq(var + 1e-5f);

    float po = 0.0f;
#pragma unroll
    for (int f = 0; f < 64; f++) {
        const float v  = w0 * E[lane][f] + w1 * E[lane][64 + f];
        const float nv = (v - mu) * rstd * ln_gamma[f] + ln_beta[f];
        po += nv * Wo[f];
    }
    out[p] = po + bo[0];
}

extern "C" void kernel_launch(void* const* d_in, const int* in_sizes, int n_in,
                              void* d_out, int out_size, void* d_ws, size_t ws_size,
                              hipStream_t stream) {
    (void)in_sizes; (void)n_in; (void)d_ws; (void)ws_size; (void)out_size;
    const float* coords     = (const float*)d_in[0];
    const float* cost_mat   = (const float*)d_in[1];
    const float* log_scale  = (const float*)d_in[2];
    const float* W1         = (const float*)d_in[3];
    const float* b1         = (const float*)d_in[4];
    const float* W2         = (const float*)d_in[5];
    const float* b2         = (const float*)d_in[6];
    const float* film_gamma = (const float*)d_in[7];
    const float* film_beta  = (const float*)d_in[8];
    const float* Wg1        = (const float*)d_in[9];
    const float* bg1        = (const float*)d_in[10];
    const float* Wg2        = (const float*)d_in[11];
    const float* bg2        = (const float*)d_in[12];
    const float* gate_temp  = (const float*)d_in[13];
    const float* ln_gamma   = (const float*)d_in[14];
    const float* ln_beta    = (const float*)d_in[15];
    const float* Wo         = (const float*)d_in[16];
    const float* bo         = (const float*)d_in[17];
    float* out = (float*)d_out;

    const int pixels_per_block = WAVES * 32;     // 64
    dim3 grid(NPIX / pixels_per_block);          // 8192
    dim3 block(THREADS);                         // 64
    GatingNeuralAdaptiveBias_kernel<<<grid, block, 0, stream>>>(
        coords, cost_mat, log_scale, W1, b1, W2, b2, film_gamma, film_beta,
        Wg1, bg1, Wg2, bg2, gate_temp, ln_gamma, ln_beta, Wo, bo, out);
}